// CorrelationMatrixLoss_70961449664956
// MI455X (gfx1250) — compile-verified
//
#include <hip/hip_runtime.h>
#include <hip/hip_bf16.h>

typedef __attribute__((ext_vector_type(2))) float v2f;
typedef __attribute__((ext_vector_type(8))) float v8f;

#define DIM 128
#define ROWS_PER_BLOCK 64
#define LDS_STRIDE 160      // 128 + 32 pad: half-wave fragment reads hit disjoint bank ranges
#define BLKTHREADS 288      // 9 wave32s -> 36 lower-triangle tiles at exactly 4 tiles/wave

// LDS byte offset of a __shared__ pointer (addrspace(3) is 32-bit on AMDGPU)
__device__ __forceinline__ unsigned lds_addr_of(const void* p) {
    return (unsigned)(unsigned long long)
        ((__attribute__((address_space(3))) const void*)p);
}

// ---------------- zero workspace ----------------
__global__ void zero_ws_kernel(float* __restrict__ ws, int n) {
    int i = blockIdx.x * blockDim.x + threadIdx.x;
    if (i < n) ws[i] = 0.0f;
}

// async-stage one 64x128 fp32 block (2048 float4) straight into padded LDS
__device__ __forceinline__ void stage_async(const float4* __restrict__ src,
                                            unsigned ldsDst, int tid) {
    for (int f = tid; f < ROWS_PER_BLOCK * (DIM / 4); f += BLKTHREADS) {
        int row = f >> 5;                 // 32 float4 per row
        int c4  = f & 31;
        unsigned dst = ldsDst + 4u * (unsigned)(row * LDS_STRIDE + c4 * 4);
        unsigned long long ga = (unsigned long long)(src + f);
        asm volatile("global_load_async_to_lds_b128 %0, %1, off"
                     :: "v"(dst), "v"(ga) : "memory");
    }
}

// ---- fused Gram (WMMA f32 16x16x4, lower-triangle tiles, double-buffered) + colsums ----
__global__ __launch_bounds__(BLKTHREADS)
void gram_colsum_kernel(const float* __restrict__ E, float* __restrict__ S,
                        float* __restrict__ colsum, int N) {
    __shared__ float lds[2][ROWS_PER_BLOCK * LDS_STRIDE];
    const int tid  = threadIdx.x;
    const int wave = tid >> 5;            // 0..8
    const int lane = tid & 31;
    const int m    = lane & 15;
    const int rsel = (lane < 16) ? 0 : 2; // A/B frag: VGPR0 holds K=0 (lo half) / K=2 (hi half)

    // 36 lower-triangle 16x16 tiles (j<=i): wave w owns tiles 4w..4w+3 (uniform, no guards)
    int Is[4], Js[4];
#pragma unroll
    for (int u = 0; u < 4; ++u) {
        int t = wave * 4 + u;
        int i = (int)floorf((sqrtf(8.0f * (float)t + 1.0f) - 1.0f) * 0.5f + 1e-4f);
        Is[u] = i;
        Js[u] = t - (i * (i + 1)) / 2;
    }

    v8f acc[4] = {};
    float csum = 0.0f;

    const unsigned base0 = lds_addr_of(&lds[0][0]);
    const unsigned base1 = lds_addr_of(&lds[1][0]);
    const int nBlocks = N / ROWS_PER_BLOCK;

    int blk = blockIdx.x;
    if (blk < nBlocks)
        stage_async((const float4*)(E + (size_t)blk * ROWS_PER_BLOCK * DIM), base0, tid);

    int parity = 0;
    for (; blk < nBlocks; blk += gridDim.x, parity ^= 1) {
        asm volatile("s_wait_asynccnt 0x0" ::: "memory");  // buf[parity] staged (own wave)
        __syncthreads();                                   // ..and globally; prev compute done
        int nxt = blk + gridDim.x;
        if (nxt < nBlocks)                                 // overlap next stage with compute
            stage_async((const float4*)(E + (size_t)nxt * ROWS_PER_BLOCK * DIM),
                        parity ? base0 : base1, tid);

        const float* buf = lds[parity];

        // column partial sums: 2 threads per column, 32 rows each (waves 0-7)
        if (tid < 256) {
            int col = tid & 127;
            int r0  = (tid >> 7) * 32;
            float s = 0.0f;
            for (int r = r0; r < r0 + 32; ++r) s += buf[r * LDS_STRIDE + col];
            csum += s;
        }

        // rank-4 updates over this wave's 4 tiles — branch-free hot loop
        for (int k0 = 0; k0 < ROWS_PER_BLOCK; k0 += 4) {
#pragma unroll
            for (int u = 0; u < 4; ++u) {
                v2f a, b;
                a.x = buf[(k0 + rsel    ) * LDS_STRIDE + Is[u] * 16 + m];
                a.y = buf[(k0 + rsel + 1) * LDS_STRIDE + Is[u] * 16 + m];
                b.x = buf[(k0 + rsel    ) * LDS_STRIDE + Js[u] * 16 + m];
                b.y = buf[(k0 + rsel + 1) * LDS_STRIDE + Js[u] * 16 + m];
                acc[u] = __builtin_amdgcn_wmma_f32_16x16x4_f32(
                    false, a, false, b, (short)0, acc[u], false, false);
            }
        }
    }

    if (tid < 256) atomicAdd(&colsum[tid & 127], csum);

    // flush: C/D layout -> VGPR r: lanes 0-15 M=r, lanes 16-31 M=r+8
    const int mbase = (lane < 16) ? 0 : 8;
    const int jin   = lane & 15;
#pragma unroll
    for (int u = 0; u < 4; ++u) {
#pragma unroll
        for (int r = 0; r < 8; ++r) {
            int gi = Is[u] * 16 + mbase + r;
            int gj = Js[u] * 16 + jin;
            atomicAdd(&S[gi * DIM + gj], acc[u][r]);
        }
    }
}

// ---------------- triplet loss: one wave per triplet, L2-resident gathers ----------------
__global__ __launch_bounds__(256)
void triplet_kernel(const float* __restrict__ E, const long long* __restrict__ trip,
                    float* __restrict__ accOut, int T) {
    const int lane   = threadIdx.x & 31;
    const int waveId = (blockIdx.x * blockDim.x + threadIdx.x) >> 5;
    const int nWaves = (gridDim.x * blockDim.x) >> 5;

    float wsum = 0.0f;
    for (int t = waveId; t < T; t += nWaves) {
        long long ip = trip[3LL * t + 0];
        long long iq = trip[3LL * t + 1];
        long long ir = trip[3LL * t + 2];
        const float4* P = (const float4*)(E + ip * DIM);
        const float4* Q = (const float4*)(E + iq * DIM);
        const float4* R = (const float4*)(E + ir * DIM);
        float4 p = P[lane], q = Q[lane], r = R[lane];

        float d = 0.0f, dx;
        dx = p.x - q.x; d += dx * dx;  dx = p.x - r.x; d -= dx * dx;
        dx = p.y - q.y; d += dx * dx;  dx = p.y - r.y; d -= dx * dx;
        dx = p.z - q.z; d += dx * dx;  dx = p.z - r.z; d -= dx * dx;
        dx = p.w - q.w; d += dx * dx;  dx = p.w - r.w; d -= dx * dx;

#pragma unroll
        for (int off = 16; off >= 1; off >>= 1) d += __shfl_xor(d, off, 32);
        if (lane == 0) wsum += fmaxf(d + 1.0f, 0.0f);   // MARGIN = 1.0
    }
    if (lane == 0) atomicAdd(accOut, wsum);
}

// ---------------- finalize: center Gram -> corr^2 tril sum, combine losses ----------------
__global__ __launch_bounds__(128)
void finalize_kernel(const float* __restrict__ S, const float* __restrict__ colsum,
                     const float* __restrict__ tripAcc, float* __restrict__ out,
                     int N, int T) {
    __shared__ float meanS[DIM], varS[DIM], red[DIM];
    const int i = threadIdx.x;
    const float fN = (float)N;
    float mean = colsum[i] / fN;
    float var  = (S[i * DIM + i] - fN * mean * mean) / (fN - 1.0f);
    meanS[i] = mean;
    varS[i]  = var;
    __syncthreads();

    float part = 0.0f;
    for (int j = 0; j < i; ++j) {
        float cov = (S[i * DIM + j] - fN * meanS[i] * meanS[j]) / (fN - 1.0f);
        part += cov * cov / (varS[i] * varS[j]);
    }
    red[i] = part;
    __syncthreads();
    for (int s = 64; s >= 1; s >>= 1) {
        if (i < s) red[i] += red[i + s];
        __syncthreads();
    }
    if (i == 0) {
        float corr = red[0] / (DIM * (DIM - 1) / 2.0f);
        out[0] = tripAcc[0] / (float)T + 0.1f * corr;
    }
}

extern "C" void kernel_launch(void* const* d_in, const int* in_sizes, int n_in,
                              void* d_out, int out_size, void* d_ws, size_t ws_size,
                              hipStream_t stream) {
    const float*     E    = (const float*)d_in[0];
    const long long* trip = (const long long*)d_in[1];
    const int N = in_sizes[0] / DIM;      // 262144 (multiple of 64)
    const int T = in_sizes[1] / 3;        // 262144

    float* ws      = (float*)d_ws;
    float* S       = ws;            // 128*128 Gram sums (lower triangle + diag valid)
    float* colsum  = ws + 16384;    // 128 column sums
    float* tripAcc = ws + 16512;    // 1 triplet-loss accumulator
    const int nz = 16513;

    zero_ws_kernel<<<(nz + 255) / 256, 256, 0, stream>>>(ws, nz);
    gram_colsum_kernel<<<256, BLKTHREADS, 0, stream>>>(E, S, colsum, N);
    triplet_kernel<<<2048, 256, 0, stream>>>(E, trip, tripAcc, T);
    finalize_kernel<<<1, 128, 0, stream>>>(S, colsum, tripAcc, (float*)d_out, N, T);
}